// MultiHeadAttention_10453950398994
// MI455X (gfx1250) — compile-verified
//
#include <hip/hip_runtime.h>

// MI455X (gfx1250, wave32) fused MHA.
//   Kernel 1: K,V projections -> head-major [B,H,S,Dh] in workspace (fp32 WMMA GEMM).
//   Kernel 2: flash-style attention with fused Q projection; scores never touch HBM.
// Matrix math: V_WMMA_F32_16X16X4_F32 (exact fp32, matches fp32 reference).
// Staging: GLOBAL_LOAD_ASYNC_TO_LDS_B32/B64 (ASYNCcnt) with pair-interleaved
// B-operand layout so every WMMA B operand is a single conflict-free ds_load_b64.

typedef float v2f __attribute__((ext_vector_type(2)));
typedef float v8f __attribute__((ext_vector_type(8)));

#define Bdim 4
#define Sdim 1024
#define Hdim 8
#define DH   64
#define DM   512
#define PAD  68    // A-tile row pitch (floats): bank = 4*row % 64 -> conflict-free A reads
#define BP   160   // B-tile kp-row pitch (floats): 128 data + pad; 160%64==32 -> halves use
                   // disjoint bank sets on ds_load_b64 B reads

static __device__ __forceinline__ v8f wmma4(v2f a, v2f b, v8f c) {
  // (neg_a, A, neg_b, B, c_mod, C, reuse_a, reuse_b)
  return __builtin_amdgcn_wmma_f32_16x16x4_f32(false, a, false, b, (short)0, c, false, false);
}

// Async global->LDS DMA (CDNA5). VDST = per-lane LDS byte offset, VADDR = 64-bit address.
static __device__ __forceinline__ void async_b64(const float* lds, const float* g) {
  unsigned off = (unsigned)(unsigned long long)lds;
  asm volatile("global_load_async_to_lds_b64 %0, %1, off" :: "v"(off), "v"(g) : "memory");
}
static __device__ __forceinline__ void async_b32(const float* lds, const float* g) {
  unsigned off = (unsigned)(unsigned long long)lds;
  asm volatile("global_load_async_to_lds_b32 %0, %1, off" :: "v"(off), "v"(g) : "memory");
}
static __device__ __forceinline__ void wait_async0() {
  asm volatile("s_wait_asynccnt 0" ::: "memory");
}

// B operand read: pair-interleaved tile, rows indexed by kp = k/2.
// lanes 0-15 need pair kk/2 (K = kk,kk+1); lanes 16-31 pair kk/2+1 (K = kk+2,kk+3).
static __device__ __forceinline__ v2f ldB(const float* base, int kk, int hi, int n) {
  return *(const v2f*)&base[((kk >> 1) + hi) * BP + 2 * n];
}

// ---------------------------------------------------------------------------
// Kernel 1: C = X @ W for (keys,Wk) and (values,Wv), stored as [B,H,S,Dh].
// Block tile 128x64, 8 waves, each wave 16x64 (4 accumulators), K chunks of 64.
// ---------------------------------------------------------------------------
__global__ __launch_bounds__(256) void proj_kv_kernel(
    const float* __restrict__ keys,   const float* __restrict__ Wk,
    const float* __restrict__ values, const float* __restrict__ Wv,
    float* __restrict__ Kb, float* __restrict__ Vb)
{
  __shared__ float Xs[128 * PAD];   // A tile, row-major
  __shared__ float Ws2[32 * BP];    // B tile, pair-interleaved

  const int blk = blockIdx.x;
  const int mat = blk >> 8;          // 0 = K, 1 = V
  const int rb  = (blk >> 3) & 31;   // 128-row block
  const int cb  = blk & 7;           // 64-col block

  const float* X = mat ? values : keys;
  const float* W = mat ? Wv : Wk;
  float*       O = mat ? Vb : Kb;

  const int tid    = threadIdx.x;
  const int wave   = tid >> 5;
  const int lane   = tid & 31;
  const int laneLo = lane & 15;
  const int hi     = lane >> 4;
  const int m0     = wave * 16;
  const int m0g    = rb * 128;
  const int n0g    = cb * 64;

  v8f acc[4] = {};

  for (int kc = 0; kc < 8; ++kc) {
    const int k0 = kc * 64;
    // X tile 128x64: contiguous 8B per element-pair -> async b64
    #pragma unroll
    for (int i = 0; i < 16; ++i) {
      int e = tid + i * 256, row = e >> 5, c2 = (e & 31) * 2;
      async_b64(&Xs[row * PAD + c2], X + (size_t)(m0g + row) * DM + k0 + c2);
    }
    // W tile 64x64 -> pair-interleaved: pair spans two k-rows -> 2x async b32
    #pragma unroll
    for (int i = 0; i < 8; ++i) {
      int e = tid + i * 256, n = e & 63, kp = e >> 6;   // kp in [0,32)
      const float* g = W + (size_t)(k0 + 2 * kp) * DM + n0g + n;
      const float* l = &Ws2[kp * BP + 2 * n];
      async_b32(l, g);
      async_b32(l + 1, g + DM);
    }
    wait_async0();
    __syncthreads();
    #pragma unroll
    for (int kk = 0; kk < 64; kk += 4) {
      v2f a = *(const v2f*)&Xs[(m0 + laneLo) * PAD + kk + hi * 2];
      #pragma unroll
      for (int t = 0; t < 4; ++t)
        acc[t] = wmma4(a, ldB(Ws2, kk, hi, t * 16 + laneLo), acc[t]);
    }
    __syncthreads();
  }

  // Store to [B,H,S,Dh]; adjacent lanes -> adjacent dh (coalesced).
  #pragma unroll
  for (int r = 0; r < 8; ++r) {
    const int mrow = m0g + m0 + r + hi * 8;
    const int b = mrow >> 10, s = mrow & 1023;
    #pragma unroll
    for (int t = 0; t < 4; ++t) {
      const int n = n0g + t * 16 + laneLo;
      const int h = n >> 6, dh = n & 63;
      O[((size_t)(b * Hdim + h) * Sdim + s) * DH + dh] = acc[t][r];
    }
  }
}

// ---------------------------------------------------------------------------
// Kernel 2: per-(b,h,q-chunk) flash attention with fused Q projection.
// 256 threads = 8 waves; wave handles 16 q rows; block handles 128 q rows.
// ---------------------------------------------------------------------------
__global__ __launch_bounds__(256) void attn_kernel(
    const float* __restrict__ queries, const float* __restrict__ Wq,
    const float* __restrict__ Kb, const float* __restrict__ Vb,
    float* __restrict__ out)
{
  __shared__ float Qs[128 * PAD];   // scaled Q tile (A layout), persistent
  __shared__ float SB[64 * BP];     // phase A: X staging (first 128*PAD);
                                    // phase B: [0,32*BP) = K^T tile, [32*BP,64*BP) = V tile
  __shared__ float SW[32 * BP];     // Wq tile, pair-interleaved
  __shared__ float Ps[8 * 16 * PAD];// per-wave P staging (C layout -> A layout)

  const int blk = blockIdx.x;
  const int qc = blk & 7, h = (blk >> 3) & 7, b = blk >> 6;

  const int tid    = threadIdx.x;
  const int wave   = tid >> 5;
  const int lane   = tid & 31;
  const int laneLo = lane & 15;
  const int hi     = lane >> 4;
  const int m0     = wave * 16;

  // ---- Phase A: Q tile projection: queries[b, qc*128..+128, :] @ Wq[:, h*64..+64]
  {
    v8f qa[4] = {};
    for (int kc = 0; kc < 8; ++kc) {
      const int k0 = kc * 64;
      #pragma unroll
      for (int i = 0; i < 16; ++i) {
        int e = tid + i * 256, row = e >> 5, c2 = (e & 31) * 2;
        async_b64(&SB[row * PAD + c2],
                  queries + ((size_t)b * Sdim + qc * 128 + row) * DM + k0 + c2);
      }
      #pragma unroll
      for (int i = 0; i < 8; ++i) {
        int e = tid + i * 256, n = e & 63, kp = e >> 6;
        const float* g = Wq + (size_t)(k0 + 2 * kp) * DM + h * 64 + n;
        const float* l = &SW[kp * BP + 2 * n];
        async_b32(l, g);
        async_b32(l + 1, g + DM);
      }
      wait_async0();
      __syncthreads();
      #pragma unroll
      for (int kk = 0; kk < 64; kk += 4) {
        v2f a = *(const v2f*)&SB[(m0 + laneLo) * PAD + kk + hi * 2];
        #pragma unroll
        for (int t = 0; t < 4; ++t)
          qa[t] = wmma4(a, ldB(SW, kk, hi, t * 16 + laneLo), qa[t]);
      }
      __syncthreads();
    }
    // Pre-scale by 1/sqrt(HEAD_DIM)=0.125 and park in LDS (A-layout row-major).
    #pragma unroll
    for (int r = 0; r < 8; ++r)
      #pragma unroll
      for (int t = 0; t < 4; ++t)
        Qs[(m0 + r + hi * 8) * PAD + t * 16 + laneLo] = qa[t][r] * 0.125f;
  }
  __syncthreads();

  // ---- Phase B: stream 64-key blocks with online softmax
  v8f o[4] = {};
  float mr[8], lr[8];
  #pragma unroll
  for (int r = 0; r < 8; ++r) { mr[r] = -3.0e38f; lr[r] = 0.0f; }

  float* Pw = Ps + wave * 16 * PAD;
  float* Ks = SB;             // K^T tile, pair-interleaved over dh
  float* Vs = SB + 32 * BP;   // V tile, pair-interleaved over key
  const float* Kbh = Kb + (size_t)(b * Hdim + h) * Sdim * DH;
  const float* Vbh = Vb + (size_t)(b * Hdim + h) * Sdim * DH;

  for (int kb = 0; kb < 16; ++kb) {
    const int key0 = kb * 64;
    // K^T tile: pair {K[key][2dhp], K[key][2dhp+1]} is contiguous 8B -> async b64
    #pragma unroll
    for (int i = 0; i < 8; ++i) {
      int e = tid + i * 256, dhp = e & 31, key = e >> 5;
      async_b64(&Ks[dhp * BP + 2 * key], Kbh + (size_t)(key0 + key) * DH + 2 * dhp);
    }
    // V tile: pair spans two key-rows -> 2x async b32
    #pragma unroll
    for (int i = 0; i < 8; ++i) {
      int e = tid + i * 256, dh = e & 63, kp = e >> 6;
      const float* g = Vbh + (size_t)(key0 + 2 * kp) * DH + dh;
      const float* l = &Vs[kp * BP + 2 * dh];
      async_b32(l, g);
      async_b32(l + 1, g + DH);
    }
    wait_async0();
    __syncthreads();

    // Scores S = Qs (16xDh) x K^T (Dh x 64)
    v8f sc[4] = {};
    #pragma unroll
    for (int kk = 0; kk < 64; kk += 4) {
      v2f a = *(const v2f*)&Qs[(m0 + laneLo) * PAD + kk + hi * 2];
      #pragma unroll
      for (int t = 0; t < 4; ++t)
        sc[t] = wmma4(a, ldB(Ks, kk, hi, t * 16 + laneLo), sc[t]);
    }

    // Online softmax (row lives across one 16-lane half in C layout)
    #pragma unroll
    for (int r = 0; r < 8; ++r) {
      float v = fmaxf(fmaxf(sc[0][r], sc[1][r]), fmaxf(sc[2][r], sc[3][r]));
      v = fmaxf(v, __shfl_xor(v, 1, 32));
      v = fmaxf(v, __shfl_xor(v, 2, 32));
      v = fmaxf(v, __shfl_xor(v, 4, 32));
      v = fmaxf(v, __shfl_xor(v, 8, 32));
      const float mnew = fmaxf(mr[r], v);
      const float corr = __expf(mr[r] - mnew);
      mr[r] = mnew;
      lr[r] *= corr;
      float rs = 0.0f;
      #pragma unroll
      for (int t = 0; t < 4; ++t) {
        float p = __expf(sc[t][r] - mnew);
        sc[t][r] = p;
        rs += p;
        o[t][r] *= corr;
      }
      rs += __shfl_xor(rs, 1, 32);
      rs += __shfl_xor(rs, 2, 32);
      rs += __shfl_xor(rs, 4, 32);
      rs += __shfl_xor(rs, 8, 32);
      lr[r] += rs;
    }

    // Re-stage P (C layout -> A layout) through wave-private LDS.
    #pragma unroll
    for (int r = 0; r < 8; ++r)
      #pragma unroll
      for (int t = 0; t < 4; ++t)
        Pw[(r + hi * 8) * PAD + t * 16 + laneLo] = sc[t][r];
    asm volatile("s_wait_dscnt 0" ::: "memory");  // wave-local LDS visibility

    // O += P (16x64) x V (64xDh)
    #pragma unroll
    for (int kk = 0; kk < 64; kk += 4) {
      v2f a = *(const v2f*)&Pw[laneLo * PAD + kk + hi * 2];
      #pragma unroll
      for (int t = 0; t < 4; ++t)
        o[t] = wmma4(a, ldB(Vs, kk, hi, t * 16 + laneLo), o[t]);
    }
    __syncthreads();
  }

  // Epilogue: out[b, q, h*64 + n] = O / l
  #pragma unroll
  for (int r = 0; r < 8; ++r) {
    const int q = qc * 128 + m0 + r + hi * 8;
    const float inv = 1.0f / lr[r];
    #pragma unroll
    for (int t = 0; t < 4; ++t)
      out[((size_t)b * Sdim + q) * DM + h * DH + t * 16 + laneLo] = o[t][r] * inv;
  }
}

// ---------------------------------------------------------------------------
extern "C" void kernel_launch(void* const* d_in, const int* in_sizes, int n_in,
                              void* d_out, int out_size, void* d_ws, size_t ws_size,
                              hipStream_t stream) {
  (void)in_sizes; (void)n_in; (void)out_size; (void)ws_size;
  const float* queries = (const float*)d_in[0];
  const float* keys    = (const float*)d_in[1];
  const float* values  = (const float*)d_in[2];
  // d_in[3] = mask: reference masking is a no-op (bug preserved) -> unused.
  const float* Wq = (const float*)d_in[4];
  const float* Wk = (const float*)d_in[5];
  const float* Wv = (const float*)d_in[6];
  float* out = (float*)d_out;

  // Workspace: K and V in [B,H,S,Dh] fp32 (8 MB each, 16 MB total).
  float* Kbuf = (float*)d_ws;
  float* Vbuf = Kbuf + (size_t)Bdim * Hdim * Sdim * DH;

  proj_kv_kernel<<<512, 256, 0, stream>>>(keys, Wk, values, Wv, Kbuf, Vbuf);
  attn_kernel<<<256, 256, 0, stream>>>(queries, Wq, Kbuf, Vbuf, out);
}